// GATProcessor_15685220565800
// MI455X (gfx1250) — compile-verified
//
#include <hip/hip_runtime.h>

// GAT (3 layers, heads=1) for MI455X / gfx1250.
// GEMM h = x @ W uses V_WMMA_F32_16X16X4_F32 (exact f32 tensor math),
// software-pipelined: 8 B-fragments per k-step loaded as ds_load_b64 into
// distinct VGPRs, next k-step prefetched under the 8-WMMA chain.
// Edge softmax/aggregation is memory-bound; coalesced float4 gathers and
// L2-resident f32 atomics.

typedef float v2f __attribute__((ext_vector_type(2)));
typedef float v8f __attribute__((ext_vector_type(8)));

#define CH 128
#define NEG_SLOPE 0.2f
#define SM_EPS 1e-16f

// order-preserving float <-> uint bijection for atomicMax-based segment max
__device__ __forceinline__ unsigned enc_f32(float f) {
  unsigned ib = __float_as_uint(f);
  return (ib & 0x80000000u) ? ~ib : (ib | 0x80000000u);
}
__device__ __forceinline__ float dec_f32(unsigned u) {
  return __uint_as_float((u & 0x80000000u) ? (u & 0x7FFFFFFFu) : ~u);
}

// ---------------------------------------------------------------------------
// GEMM: H[N,128] = X[N,128] @ W[128,128] via WMMA f32 16x16x4.
// Block = 256 threads (8 waves). Each wave owns 16 rows x 128 cols
// (8 accumulator tiles of 16x16). W^T staged in LDS with padded stride 132
// (132 % 64 == 4 -> all 32 lanes hit distinct bank pairs on b64 reads).
// A fragment (16x4 f32, 2 VGPRs/lane): lanes 0-15 hold K=k,k+1 of row M=lane;
// lanes 16-31 hold K=k+2,k+3 of row M=lane-16.  B mirrors this per-K split.
// ---------------------------------------------------------------------------
#define WT_PAD 132

__global__ void gat_gemm_wmma(const float* __restrict__ X,
                              const float* __restrict__ W,
                              float* __restrict__ H, int n_nodes) {
  extern __shared__ float wt[];  // [128][WT_PAD] : wt[n][k] = W[k][n]
  const int tid = threadIdx.x;
  for (int i = tid; i < CH * CH; i += 256) {
    int k = i >> 7;       // row of W
    int n = i & (CH - 1); // col of W
    wt[n * WT_PAD + k] = W[i];
  }
  __syncthreads();

  const int wave = tid >> 5;
  const int lane = tid & 31;
  const int row0 = blockIdx.x * 128 + wave * 16;
  const int mrow = lane & 15;
  const int koff = (lane >> 4) * 2;  // 0 for lanes 0-15, 2 for lanes 16-31

  int rload = row0 + mrow;
  if (rload >= n_nodes) rload = n_nodes - 1;  // clamp; stores are guarded
  const float* xrow = X + (size_t)rload * CH + koff;
  const float* wrow = wt + mrow * WT_PAD + koff;

  v8f acc[8];
#pragma unroll
  for (int t = 0; t < 8; ++t) acc[t] = {};

  // prologue: fragments for k4 = 0
  v2f a = *(const v2f*)(xrow);
  v2f b[8];
#pragma unroll
  for (int t = 0; t < 8; ++t)
    b[t] = *(const v2f*)(wrow + t * 16 * WT_PAD);

  for (int k4 = 0; k4 < CH; k4 += 4) {
    const int kn = k4 + 4;
    v2f an = {};
    v2f bn[8];
#pragma unroll
    for (int t = 0; t < 8; ++t) bn[t] = {};
    if (kn < CH) {
      an = *(const v2f*)(xrow + kn);
#pragma unroll
      for (int t = 0; t < 8; ++t)
        bn[t] = *(const v2f*)(wrow + t * 16 * WT_PAD + kn);
    }
#pragma unroll
    for (int t = 0; t < 8; ++t)
      acc[t] = __builtin_amdgcn_wmma_f32_16x16x4_f32(
          false, a, false, b[t], (short)0, acc[t], false, false);
    a = an;
#pragma unroll
    for (int t = 0; t < 8; ++t) b[t] = bn[t];
  }

  // C/D layout: VGPR r -> lanes 0-15: (M=r, N=lane); lanes 16-31: (M=r+8)
  const int rhi = (lane >> 4) * 8;
#pragma unroll
  for (int t = 0; t < 8; ++t) {
    const int col = t * 16 + (lane & 15);
#pragma unroll
    for (int r = 0; r < 8; ++r) {
      int row = row0 + r + rhi;
      if (row < n_nodes) H[(size_t)row * CH + col] = acc[t][r];
    }
  }
}

// ---------------------------------------------------------------------------
// Per-node attention scores: s[i] = h[i] . a_src, d[i] = h[i] . a_dst
// One wave per node; float4 loads; wave32 shuffle reduction.
// ---------------------------------------------------------------------------
__global__ void gat_node_scores(const float* __restrict__ H,
                                const float* __restrict__ a_src,
                                const float* __restrict__ a_dst,
                                float* __restrict__ s, float* __restrict__ d,
                                int n_nodes) {
  const int wave = threadIdx.x >> 5;
  const int lane = threadIdx.x & 31;
  const int node = blockIdx.x * 8 + wave;
  if (node >= n_nodes) return;
  float4 h4 = ((const float4*)(H + (size_t)node * CH))[lane];
  float4 a4 = ((const float4*)a_src)[lane];
  float4 b4 = ((const float4*)a_dst)[lane];
  float ss = h4.x * a4.x + h4.y * a4.y + h4.z * a4.z + h4.w * a4.w;
  float dd = h4.x * b4.x + h4.y * b4.y + h4.z * b4.z + h4.w * b4.w;
#pragma unroll
  for (int o = 16; o > 0; o >>= 1) {
    ss += __shfl_xor(ss, o, 32);
    dd += __shfl_xor(dd, o, 32);
  }
  if (lane == 0) {
    s[node] = ss;
    d[node] = dd;
  }
}

// ---------------------------------------------------------------------------
// Per-layer scratch init: agg = 0, emax = 0 (below any encoded finite), denom = 0
// ---------------------------------------------------------------------------
__global__ void gat_init(float* __restrict__ agg, unsigned* __restrict__ emax,
                         float* __restrict__ denom, int n_nodes) {
  size_t i = (size_t)blockIdx.x * blockDim.x + threadIdx.x;
  size_t total = (size_t)n_nodes * CH;
  if (i < total) agg[i] = 0.0f;
  if (i < (size_t)n_nodes) {
    emax[i] = 0u;  // below encode() of any finite float
    denom[i] = 0.0f;
  }
}

// ---------------------------------------------------------------------------
// Pass 1 over edges: e = leakyrelu(s[src] + d[dst]); segment max by dst.
// Edge ids >= E are self-loops (src = dst = id - E).
// ---------------------------------------------------------------------------
__global__ void gat_edge_max(const int* __restrict__ src,
                             const int* __restrict__ dst,
                             const float* __restrict__ s,
                             const float* __restrict__ d,
                             float* __restrict__ ework,
                             unsigned* __restrict__ emax, int n_edges,
                             int n_nodes) {
  int e = blockIdx.x * blockDim.x + threadIdx.x;
  int tot = n_edges + n_nodes;
  if (e >= tot) return;
  int u = (e < n_edges) ? src[e] : (e - n_edges);
  int v = (e < n_edges) ? dst[e] : (e - n_edges);
  float val = s[u] + d[v];
  val = (val > 0.0f) ? val : NEG_SLOPE * val;
  ework[e] = val;
  atomicMax(&emax[v], enc_f32(val));
}

// ---------------------------------------------------------------------------
// Pass 2: w = exp(e - max[dst]); segment sum by dst.
// ---------------------------------------------------------------------------
__global__ void gat_edge_expsum(const int* __restrict__ dst,
                                float* __restrict__ ework,
                                const unsigned* __restrict__ emax,
                                float* __restrict__ denom, int n_edges,
                                int n_nodes) {
  int e = blockIdx.x * blockDim.x + threadIdx.x;
  int tot = n_edges + n_nodes;
  if (e >= tot) return;
  int v = (e < n_edges) ? dst[e] : (e - n_edges);
  float w = __builtin_expf(ework[e] - dec_f32(emax[v]));
  ework[e] = w;
  atomicAdd(&denom[v], w);
}

// ---------------------------------------------------------------------------
// Pass 3: agg[dst] += (w / (denom[dst]+eps)) * h[src].  One wave per edge:
// each lane handles 4 channels (float4 gather + 4 f32 atomics).
// ---------------------------------------------------------------------------
__global__ void gat_edge_aggregate(const int* __restrict__ src,
                                   const int* __restrict__ dst,
                                   const float* __restrict__ ework,
                                   const float* __restrict__ denom,
                                   const float* __restrict__ H,
                                   float* __restrict__ agg, int n_edges,
                                   int n_nodes) {
  int wid = (int)(((size_t)blockIdx.x * blockDim.x + threadIdx.x) >> 5);
  int lane = threadIdx.x & 31;
  int tot = n_edges + n_nodes;
  if (wid >= tot) return;
  int u = (wid < n_edges) ? src[wid] : (wid - n_edges);
  int v = (wid < n_edges) ? dst[wid] : (wid - n_edges);
  float alpha = ework[wid] / (denom[v] + SM_EPS);
  float4 h4 = ((const float4*)(H + (size_t)u * CH))[lane];
  float* ap = agg + (size_t)v * CH + lane * 4;
  atomicAdd(ap + 0, alpha * h4.x);
  atomicAdd(ap + 1, alpha * h4.y);
  atomicAdd(ap + 2, alpha * h4.z);
  atomicAdd(ap + 3, alpha * h4.w);
}

// ---------------------------------------------------------------------------
// Residual update: x += sigmoid(agg + bias)
// ---------------------------------------------------------------------------
__global__ void gat_update(float* __restrict__ x, const float* __restrict__ agg,
                           const float* __restrict__ bias, int n_nodes) {
  size_t i = (size_t)blockIdx.x * blockDim.x + threadIdx.x;
  if (i >= (size_t)n_nodes * CH) return;
  float v = agg[i] + bias[i & (CH - 1)];
  x[i] += 1.0f / (1.0f + __builtin_expf(-v));
}

__global__ void gat_copy(const float* __restrict__ in, float* __restrict__ out,
                         size_t n) {
  size_t i = (size_t)blockIdx.x * blockDim.x + threadIdx.x;
  if (i < n) out[i] = in[i];
}

// ---------------------------------------------------------------------------
extern "C" void kernel_launch(void* const* d_in, const int* in_sizes, int n_in,
                              void* d_out, int out_size, void* d_ws,
                              size_t ws_size, hipStream_t stream) {
  const float* x_in = (const float*)d_in[0];
  const int* ei = (const int*)d_in[1];  // [2, E]: row0 = src, row1 = dst
  const float* Ws = (const float*)d_in[2];
  const float* att_src = (const float*)d_in[3];
  const float* att_dst = (const float*)d_in[4];
  const float* bias = (const float*)d_in[5];

  const int N = in_sizes[0] / CH;
  const int E = in_sizes[1] / 2;
  const int M = in_sizes[3] / CH;  // number of layers
  const int* src = ei;
  const int* dst = ei + E;
  const int TOT = E + N;  // edges + self loops

  // workspace carve-up (floats)
  float* ws = (float*)d_ws;
  float* h = ws;                     // N*CH
  float* agg = h + (size_t)N * CH;   // N*CH
  float* s = agg + (size_t)N * CH;   // N
  float* dsc = s + N;                // N
  float* denom = dsc + N;            // N
  unsigned* emax = (unsigned*)(denom + N);  // N
  float* ework = (float*)(emax + N);        // E+N

  float* x = (float*)d_out;  // evolving node features

  const size_t nfeat = (size_t)N * CH;
  const int blk = 256;
  const int g_feat = (int)((nfeat + blk - 1) / blk);
  const int g_gemm = (N + 127) / 128;
  const int g_node = (N + 7) / 8;
  const int g_edge = (TOT + blk - 1) / blk;
  const int g_eagg = (TOT + 7) / 8;  // one wave32 per edge
  const size_t smem = (size_t)CH * WT_PAD * sizeof(float);  // ~67.6 KB

  gat_copy<<<g_feat, blk, 0, stream>>>(x_in, x, nfeat);

  for (int m = 0; m < M; ++m) {
    const float* W = Ws + (size_t)m * CH * CH;
    const float* as_ = att_src + (size_t)m * CH;
    const float* ad_ = att_dst + (size_t)m * CH;
    const float* b_ = bias + (size_t)m * CH;

    gat_gemm_wmma<<<g_gemm, blk, smem, stream>>>(x, W, h, N);
    gat_node_scores<<<g_node, blk, 0, stream>>>(h, as_, ad_, s, dsc, N);
    gat_init<<<g_feat, blk, 0, stream>>>(agg, emax, denom, N);
    gat_edge_max<<<g_edge, blk, 0, stream>>>(src, dst, s, dsc, ework, emax, E, N);
    gat_edge_expsum<<<g_edge, blk, 0, stream>>>(dst, ework, emax, denom, E, N);
    gat_edge_aggregate<<<g_eagg, blk, 0, stream>>>(src, dst, ework, denom, h,
                                                   agg, E, N);
    gat_update<<<g_feat, blk, 0, stream>>>(x, agg, b_, N);
  }
}